// GraphCNN_3805341024422
// MI455X (gfx1250) — compile-verified
//
#include <hip/hip_runtime.h>

// MI455X (gfx1250) GraphCNN: out = relu(G @ (X @ W) + bias)
// B=2, N=16384, F=D=16.  Memory-bound: stream 1 GiB of G once (~46us at
// 23.3 TB/s).  Main GEMM runs on v_wmma_f32_16x16x4_f32 chains.  Both
// operands are fetched as single b128 per lane per 8-wide K step by
// choosing a custom (consistent) K-permutation for A and B:
//   HW K-slot (v,h) = v + 2h  <->  G column / Y row  k + 4h + v.

typedef __attribute__((ext_vector_type(2))) float v2f;
typedef __attribute__((ext_vector_type(4))) float v4f;
typedef __attribute__((ext_vector_type(8))) float v8f;

#define N_NODES 16384
#define F_DIM   16
#define B_BATCH 2
#define WAVES_PER_BLOCK 8
#define K_PER_WAVE (N_NODES / WAVES_PER_BLOCK)   // 2048

// Swizzled Y panel: Yp[b][k>>3][h][n][v] = Y[b, 8*(k>>3) + 4h + v, n]
//   flat = b*(N*16) + (k>>3)*128 + h*64 + n*4 + v     (v = 0..3)
// Lane L (h = L>>4, n = L&15) reads its four B values as one b128;
// a half-wave covers 256 contiguous bytes.

// ---------------------------------------------------------------------------
// Kernel 1: Yp = swizzle(X @ W)   (tiny: 2 MiB out, ~17 MFLOP)
// ---------------------------------------------------------------------------
__global__ __launch_bounds__(256) void precompute_xw(
    const float* __restrict__ x, const float* __restrict__ w,
    float* __restrict__ yp) {
  __shared__ float ws[F_DIM * F_DIM];
  int t = threadIdx.x;
  if (t < F_DIM * F_DIM) ws[t] = w[t];
  __syncthreads();

  int row = blockIdx.x * 256 + t;          // 0 .. B*N-1
  const float* xr = x + (size_t)row * F_DIM;
  float xv[F_DIM];
#pragma unroll
  for (int f = 0; f < F_DIM; ++f) xv[f] = xr[f];

  int b = row / N_NODES;
  int n = row - b * N_NODES;               // Y row index (K dim of main GEMM)
  // scatter base inside the swizzled panel: h=(n>>2)&1, v=n&3
  float* dst = yp + (size_t)b * N_NODES * F_DIM
                  + (size_t)(n >> 3) * 128 + ((n >> 2) & 1) * 64 + (n & 3);
#pragma unroll
  for (int d = 0; d < F_DIM; ++d) {
    float acc = 0.f;
#pragma unroll
    for (int f = 0; f < F_DIM; ++f) acc += xv[f] * ws[f * F_DIM + d];
    dst[d * 4] = acc;
  }
}

// ---------------------------------------------------------------------------
// Kernel 2: out[b, m, :] = relu( G[m, :] @ Y[b] + bias )
// One block (8 waves) per 16-row tile of G; waves split the K=16384
// reduction 8 ways; 4 independent WMMA accumulation chains per wave
// hide XDL latency; partials reduced through LDS.
// ---------------------------------------------------------------------------
__global__ __launch_bounds__(256) void graph_gemm(
    const float* __restrict__ g, const float* __restrict__ yp,
    const float* __restrict__ bias, float* __restrict__ out) {
  __shared__ float red[WAVES_PER_BLOCK][B_BATCH][8][32];  // 16 KB

  const int lane = threadIdx.x & 31;
  const int wave = threadIdx.x >> 5;
  const int half = lane >> 4;   // 0 or 1 (upper half-wave)
  const int l15  = lane & 15;
  const int m0   = blockIdx.x * 16;

  // A: lane L loads G[m0+(L&15), k+4h .. k+4h+3] as one b128 (NT stream).
  const float* ga = g + (size_t)(m0 + l15) * N_NODES + 4 * half;
  // B: swizzled panel, one b128 per batch per 8-wide K step (L2-hot).
  const float* yb0 = yp + half * 64 + l15 * 4;                   // batch 0
  const float* yb1 = yb0 + (size_t)N_NODES * F_DIM;              // batch 1

  v8f c0a = {}, c0b = {};   // batch 0: two K-chunk chains
  v8f c1a = {}, c1b = {};   // batch 1: two K-chunk chains

  const int k_begin = wave * K_PER_WAVE;
  const int k_end   = k_begin + K_PER_WAVE;

#pragma unroll 4
  for (int k = k_begin; k < k_end; k += 8) {
    v4f a  = __builtin_nontemporal_load((const v4f*)(ga + k));
    v4f b0 = *(const v4f*)(yb0 + (size_t)k * 16);   // (k>>3)*128 floats
    v4f b1 = *(const v4f*)(yb1 + (size_t)k * 16);

    v2f a01  = {a.x, a.y},   a23  = {a.z, a.w};
    v2f b0lo = {b0.x, b0.y}, b0hi = {b0.z, b0.w};
    v2f b1lo = {b1.x, b1.y}, b1hi = {b1.z, b1.w};

    // D = A*B + C   (neg_a, A, neg_b, B, c_mod, C, reuse_a, reuse_b)
    c0a = __builtin_amdgcn_wmma_f32_16x16x4_f32(false, a01, false, b0lo,
                                                (short)0, c0a, false, false);
    c1a = __builtin_amdgcn_wmma_f32_16x16x4_f32(false, a01, false, b1lo,
                                                (short)0, c1a, false, false);
    c0b = __builtin_amdgcn_wmma_f32_16x16x4_f32(false, a23, false, b0hi,
                                                (short)0, c0b, false, false);
    c1b = __builtin_amdgcn_wmma_f32_16x16x4_f32(false, a23, false, b1hi,
                                                (short)0, c1b, false, false);
  }

  v8f c0 = c0a + c0b;
  v8f c1 = c1a + c1b;

  // Spill per-wave partials to LDS, reduce across the 8 waves.
#pragma unroll
  for (int v = 0; v < 8; ++v) {
    red[wave][0][v][lane] = c0[v];
    red[wave][1][v][lane] = c1[v];
  }
  __syncthreads();

  // 512 output elements per tile (2 batches x 8 vgprs x 32 lanes);
  // 256 threads handle 2 each. No divergence (EXEC stays all-ones).
  const int t = threadIdx.x;
#pragma unroll
  for (int i = 0; i < 2; ++i) {
    int e  = t + i * 256;
    int b  = e >> 8;
    int v  = (e >> 5) & 7;
    int ln = e & 31;
    float s = 0.f;
#pragma unroll
    for (int w = 0; w < WAVES_PER_BLOCK; ++w) s += red[w][b][v][ln];
    int r   = m0 + v + 8 * (ln >> 4);   // D-layout: vgpr v -> rows v / v+8
    int col = ln & 15;
    float val = s + bias[col];
    out[((size_t)b * N_NODES + r) * F_DIM + col] = fmaxf(val, 0.f);
  }
}

// ---------------------------------------------------------------------------
extern "C" void kernel_launch(void* const* d_in, const int* in_sizes, int n_in,
                              void* d_out, int out_size, void* d_ws,
                              size_t ws_size, hipStream_t stream) {
  const float* x    = (const float*)d_in[0];  // (B, N, F)
  const float* g    = (const float*)d_in[1];  // (N, N)
  const float* w    = (const float*)d_in[2];  // (F, D)
  const float* bias = (const float*)d_in[3];  // (D,)
  float* yp  = (float*)d_ws;                  // swizzled (B, N, 16) = 2 MiB
  float* out = (float*)d_out;                 // (B, N, D)

  precompute_xw<<<(B_BATCH * N_NODES) / 256, 256, 0, stream>>>(x, w, yp);
  graph_gemm<<<N_NODES / 16, 256, 0, stream>>>(g, yp, bias, out);
}